// Prompts_4904852652177
// MI455X (gfx1250) — compile-verified
//
#include <hip/hip_runtime.h>
#include <math.h>

typedef float v2f __attribute__((ext_vector_type(2)));
typedef float v8f __attribute__((ext_vector_type(8)));

#define D_DIM         512
#define K_PROMPTS     2
#define ROWS_PER_WAVE 16
#define WAVES_PER_WG  8
#define ROWS_PER_WG   (ROWS_PER_WAVE * WAVES_PER_WG)   // 128
#define CHUNK         32
#define NCHUNK        (D_DIM / CHUNK)                  // 16
#define XT_STRIDE     36   // 16B-aligned rows, bank-conflict-free b64 reads

__global__ __launch_bounds__(256) void prompts_softmax_wmma(
    const float* __restrict__ x,
    const float* __restrict__ tf,
    const float* __restrict__ logit_scale,
    float* __restrict__ out)
{
    __shared__ float s_scale[K_PROMPTS];
    __shared__ __attribute__((aligned(16))) float s_tfn[K_PROMPTS][D_DIM];
    // per-wave double-buffered x tiles
    __shared__ __attribute__((aligned(16))) float s_xt[WAVES_PER_WG][2][ROWS_PER_WAVE][XT_STRIDE];

    const int tid  = threadIdx.x;
    const int w    = tid >> 5;    // wave id 0..7
    const int lane = tid & 31;

    // ---- per-workgroup: tfn = exp(logit_scale) * tf_row / ||tf_row|| into LDS ----
    if (w < K_PROMPTS) {
        float ss = 0.f;
        for (int i = lane; i < D_DIM; i += 32) {
            float v = tf[w * D_DIM + i];
            ss += v * v;
        }
        #pragma unroll
        for (int off = 16; off > 0; off >>= 1) ss += __shfl_xor(ss, off, 32);
        if (lane == 0) s_scale[w] = __expf(logit_scale[0]) / sqrtf(ss);
    }
    __syncthreads();
    for (int i = tid; i < K_PROMPTS * D_DIM; i += 256) {
        s_tfn[i / D_DIM][i % D_DIM] = tf[i] * s_scale[i / D_DIM];
    }
    __syncthreads();

    const int rowBase = blockIdx.x * ROWS_PER_WG + w * ROWS_PER_WAVE;
    const int r    = lane >> 1;          // staging row (2 lanes per row)
    const int cb   = (lane & 1) * 16;    // staging col base within chunk
    const int m    = lane & 15;          // fragment row index
    const int off2 = (lane >> 4) << 1;   // K sub-offset: 0 (lanes<16) or 2

    // branch-free A-fragment masking: only prompt rows 0..1 are live
    const float amask = (m < K_PROMPTS) ? 1.f : 0.f;
    const int   mrow  = m & (K_PROMPTS - 1);

    const float* xrow = x + (size_t)(rowBase + r) * D_DIM + cb;

    // ---- prologue: stage chunk 0 into buffer 0 (coalesced 128-bit loads) ----
    {
        float4 f0 = *(const float4*)(xrow + 0);
        float4 f1 = *(const float4*)(xrow + 4);
        float4 f2 = *(const float4*)(xrow + 8);
        float4 f3 = *(const float4*)(xrow + 12);
        float* dst = &s_xt[w][0][r][cb];
        *(float4*)(dst + 0)  = f0;
        *(float4*)(dst + 4)  = f1;
        *(float4*)(dst + 8)  = f2;
        *(float4*)(dst + 12) = f3;
    }
    __builtin_amdgcn_wave_barrier();   // order stores before this wave's ds reads

    v8f c = {0.f, 0.f, 0.f, 0.f, 0.f, 0.f, 0.f, 0.f};

    #pragma unroll
    for (int ci = 0; ci < NCHUNK; ++ci) {
        const int d0  = ci * CHUNK;
        const int cur = ci & 1;

        // issue next chunk's global loads early: latency hides behind WMMA
        float4 g0, g1, g2, g3;
        if (ci + 1 < NCHUNK) {
            const float* nx = xrow + d0 + CHUNK;
            g0 = *(const float4*)(nx + 0);
            g1 = *(const float4*)(nx + 4);
            g2 = *(const float4*)(nx + 8);
            g3 = *(const float4*)(nx + 12);
            if (ci + 2 < NCHUNK)
                __builtin_prefetch(xrow + d0 + 2 * CHUNK, 0, 3);
        }

        // consume current buffer: 8 x v_wmma_f32_16x16x4_f32
        #pragma unroll
        for (int kk = 0; kk < CHUNK / 4; ++kk) {
            const int col = kk * 4 + off2;
            const float2 bv = *(const float2*)&s_xt[w][cur][m][col];
            const float2 av = *(const float2*)&s_tfn[mrow][d0 + col];
            v2f b; b.x = bv.x;         b.y = bv.y;
            v2f a; a.x = av.x * amask; a.y = av.y * amask;
            // D[m][n] += sum_k A[m][k]*B[k][n]  (m=prompt, n=x-row)
            c = __builtin_amdgcn_wmma_f32_16x16x4_f32(
                    false, a, false, b, (short)0, c, false, false);
        }

        // stage next chunk into the alternate buffer
        if (ci + 1 < NCHUNK) {
            float* dst = &s_xt[w][cur ^ 1][r][cb];
            *(float4*)(dst + 0)  = g0;
            *(float4*)(dst + 4)  = g1;
            *(float4*)(dst + 8)  = g2;
            *(float4*)(dst + 12) = g3;
        }
        __builtin_amdgcn_wave_barrier();   // keep store->load order across iterations
    }

    // lanes 0..15: c[0] = logit(prompt0, row=lane), c[1] = logit(prompt1, row=lane)
    if (lane < 16) {
        const int row = rowBase + lane;
        const float l0 = c[0], l1 = c[1];
        const float mx = fmaxf(l0, l1);
        const float e0 = __expf(l0 - mx);
        const float e1 = __expf(l1 - mx);
        const float inv = 1.f / (e0 + e1);
        float2 p; p.x = e0 * inv; p.y = e1 * inv;
        *(float2*)(out + (size_t)row * 2) = p;   // coalesced 128B per 16 lanes
    }
}

extern "C" void kernel_launch(void* const* d_in, const int* in_sizes, int n_in,
                              void* d_out, int out_size, void* d_ws, size_t ws_size,
                              hipStream_t stream) {
    const float* x  = (const float*)d_in[0];   // [B, N, D] f32
    const float* tf = (const float*)d_in[1];   // [K, D]    f32
    const float* ls = (const float*)d_in[2];   // [1]       f32
    float* out = (float*)d_out;                // [B*N, K]  f32

    const int rows = in_sizes[0] / D_DIM;      // B*N = 262144
    const int grid = rows / ROWS_PER_WG;       // 2048
    prompts_softmax_wmma<<<grid, 256, 0, stream>>>(x, tf, ls, out);
}